// ScorePredictor_56968446214864
// MI455X (gfx1250) — compile-verified
//
#include <hip/hip_runtime.h>
#include <hip/hip_bf16.h>

#define F_DIM   128
#define D_DIM   256
#define NCATE   18
#define P_DIM   8
#define WAVE    32
#define WPB     8            // waves per block -> 256 threads
#define ROWPAD  132          // padded LDS row stride (floats): 4-bank shift/row

typedef float v2f  __attribute__((ext_vector_type(2)));
typedef float v8f  __attribute__((ext_vector_type(8)));
typedef float f32x4 __attribute__((ext_vector_type(4)));

__global__ __launch_bounds__(WPB * WAVE)
void score_predictor_kernel(const float* __restrict__ gnn_emb_src,
                            const float* __restrict__ gnn_emb_dst,
                            const float* __restrict__ news_pref,
                            const float* __restrict__ last_update_time,
                            const float* __restrict__ time_arr,
                            const int*   __restrict__ cate_id,
                            const int*   __restrict__ src_idx,
                            const int*   __restrict__ dst_idx,
                            float*       __restrict__ out,
                            int E)
{
    // per-wave private staging: 10 matrix rows (8 pref + target + zero) + src row
    __shared__ float lds_rows[WPB][10][ROWPAD];
    __shared__ float lds_src [WPB][F_DIM];

    const int lane = threadIdx.x & (WAVE - 1);
    const int wid  = threadIdx.x >> 5;
    const int e    = blockIdx.x * WPB + wid;
    if (e >= E) return;                         // wave-uniform: EXEC stays all-ones

    const int s  = src_idx[e];
    const int dd = dst_idx[e];
    const int c  = cate_id[s];
    const float t0 = time_arr[0];

    const float* srcRow   = gnn_emb_src + (size_t)s * D_DIM;
    const float* dstRow   = gnn_emb_dst + (size_t)dd * D_DIM;
    const float* prefBase = news_pref + ((size_t)dd * NCATE + c) * P_DIM * D_DIM;
    const float* lutBase  = last_update_time + ((size_t)dd * NCATE + c) * P_DIM;

    // ---- stage gathers into LDS (coalesced float4 per lane: 128 floats/row) ----
    *(f32x4*)(&lds_src[wid][lane * 4]) = *(const f32x4*)(srcRow + lane * 4);
#pragma unroll
    for (int r = 0; r < P_DIM; ++r)
        *(f32x4*)(&lds_rows[wid][r][lane * 4]) =
            *(const f32x4*)(prefBase + (size_t)r * D_DIM + lane * 4);
    *(f32x4*)(&lds_rows[wid][8][lane * 4]) = *(const f32x4*)(dstRow + lane * 4);
    {   f32x4 z = {0.f, 0.f, 0.f, 0.f};
        *(f32x4*)(&lds_rows[wid][9][lane * 4]) = z; }

    float lut_l = lutBase[lane & (P_DIM - 1)];  // lane p holds lut[p]

    // cross-lane LDS RAW: DS is in-order per wave; fence compiler + wait hardware
    asm volatile("s_wait_dscnt 0" ::: "memory");

    // ---- 32 x V_WMMA_F32_16X16X4_F32, K accumulation over F=128 ----
    // A 16x4 layout: lane L holds M=L&15, K = ((L>=16)?2:0) + vgpr
    // B  4x16 layout: same K pattern, N = L&15 (src broadcast across N)
    const int m    = lane & 15;
    const int rm   = (m <= 8) ? m : 9;          // rows 9..15 read the zero row
    const int koff = (lane >> 4) << 1;          // 0 or 2
    const float* arow = &lds_rows[wid][rm][koff];
    const float* brow = &lds_src[wid][koff];

    v8f acc = {};
#pragma unroll
    for (int k0 = 0; k0 < F_DIM; k0 += 4) {
        v2f a = *(const v2f*)(arow + k0);       // 8B aligned ds_load_b64
        v2f b = *(const v2f*)(brow + k0);
        acc = __builtin_amdgcn_wmma_f32_16x16x4_f32(false, a, false, b,
                                                    (short)0, acc, false, false);
    }

    // ---- finish: D[p,0]=dp[p] (lane 0, vgpr p); D[8,0]=dt (lane 16, vgpr 0) ----
    float dt_dot = __shfl(acc[0], 16, WAVE);
    float dp[P_DIM];
#pragma unroll
    for (int p = 0; p < P_DIM; ++p) dp[p] = __shfl(acc[p], 0, WAVE);

    const float inv_sqrtF = 0.08838834764831845f;   // 1/sqrt(128)
    float sraw[P_DIM];
    float mx = -3.402823466e38f;
#pragma unroll
    for (int p = 0; p < P_DIM; ++p) {
        float lut_p = __shfl(lut_l, p, WAVE);
        float w  = expf(lut_p - t0);                // exp(-(t0-lut)/TAU), TAU=1
        float sr = w * dp[p] + (1.0f - w) * dt_dot;
        sraw[p] = sr;
        mx = fmaxf(mx, sr * inv_sqrtF);
    }
    float den = 0.f, num = 0.f;
#pragma unroll
    for (int p = 0; p < P_DIM; ++p) {
        float a = expf(sraw[p] * inv_sqrtF - mx);
        den += a;
        num += a * sraw[p];
    }
    if (lane == 0) out[e] = num / den;
}

extern "C" void kernel_launch(void* const* d_in, const int* in_sizes, int n_in,
                              void* d_out, int out_size, void* d_ws, size_t ws_size,
                              hipStream_t stream) {
    const float* gsrc = (const float*)d_in[0];
    const float* gdst = (const float*)d_in[1];
    const float* pref = (const float*)d_in[2];
    const float* lut  = (const float*)d_in[3];
    const float* tim  = (const float*)d_in[4];
    const int*   cate = (const int*)d_in[5];
    const int*   sidx = (const int*)d_in[6];
    const int*   didx = (const int*)d_in[7];
    float* out = (float*)d_out;

    const int E = in_sizes[6];                  // src_idx has E entries
    const int blocks = (E + WPB - 1) / WPB;
    score_predictor_kernel<<<blocks, WPB * WAVE, 0, stream>>>(
        gsrc, gdst, pref, lut, tim, cate, sidx, didx, out, E);
}